// KalmanFilter_12498354831460
// MI455X (gfx1250) — compile-verified
//
#include <hip/hip_runtime.h>
#include <stdint.h>

#define B_ 256
#define T_ 512
#define S_ 64
#define M_ 32

typedef float v2f __attribute__((ext_vector_type(2)));
typedef float v8f __attribute__((ext_vector_type(8)));
typedef unsigned int u32x4 __attribute__((ext_vector_type(4)));
typedef int i32x4 __attribute__((ext_vector_type(4)));
typedef int i32x8 __attribute__((ext_vector_type(8)));

// V_WMMA_F32_16X16X4_F32 : D(16x16 f32) = A(16x4 f32) x B(4x16 f32) + C
__device__ __forceinline__ v8f wmma_f32_16x16x4(v2f a, v2f b, v8f c) {
  return __builtin_amdgcn_wmma_f32_16x16x4_f32(false, a, false, b, (short)0, c,
                                               false, false);
}

// ---------------------------------------------------------------------------
// Tensor Data Mover: 1-D DMA of one gain matrix K_t (2048 f32 = 8 KB) from
// global memory into LDS. D# per CDNA5 ISA 8.3/8.4: count=1, type=2("image"),
// data_size=4B, tile_dim0 = tensor_dim0 = 2048, 1 row. Groups 2/3 zero
// (tensor <= 2D). Issued once by one wave; tracked with TENSORcnt.
// ---------------------------------------------------------------------------
__device__ __forceinline__ void tdm_copy_gain(const float* __restrict__ src,
                                              unsigned lds_off) {
  uint64_t ga = (uint64_t)(uintptr_t)src;
  u32x4 g0;
  g0[0] = 1u;                                   // count=1 (valid descriptor)
  g0[1] = lds_off;                              // lds_addr (bytes)
  g0[2] = (unsigned)(ga & 0xffffffffu);         // global_addr[31:0]
  g0[3] = (unsigned)((ga >> 32) & 0x01ffffffu)  // global_addr[56:32]
          | (2u << 30);                         // type = 2
  i32x8 g1;
  g1[0] = 0x00020000;             // workgroup_mask=0, data_size=2 -> 4 bytes
  g1[1] = (int)(2048u << 16);     // tensor_dim0[15:0] @bit48 = 2048
  g1[2] = (int)(1u << 16);        // tensor_dim0 hi=0 ; tensor_dim1 = 1
  g1[3] = (int)(2048u << 16);     // tile_dim0 @[127:112] = 2048
  g1[4] = 1;                      // tile_dim1 = 1, tile_dim2 = 0
  g1[5] = 2048;                   // tensor_dim0_stride = 2048
  g1[6] = 0;
  g1[7] = 0;
  i32x4 gz = {0, 0, 0, 0};
#if defined(__clang_major__) && __clang_major__ >= 23
  i32x8 gz8 = {0, 0, 0, 0, 0, 0, 0, 0};
  __builtin_amdgcn_tensor_load_to_lds(g0, g1, gz, gz, gz8, 0);
#else
  __builtin_amdgcn_tensor_load_to_lds(g0, g1, gz, gz, 0);
#endif
}

// ---------------------------------------------------------------------------
// Kernel 1: batch-independent covariance / gain recursion.
// One workgroup runs the serial chain; K_t (64x32) stored to d_ws for all t.
// ---------------------------------------------------------------------------
__global__ __launch_bounds__(256) void kf_gains(
    const float* __restrict__ F, const float* __restrict__ H,
    const float* __restrict__ Q, const float* __restrict__ R,
    const float* __restrict__ P0, float* __restrict__ Kall) {
  __shared__ float P[S_ * S_];         // current covariance (pred or filt)
  __shared__ float Pp[S_ * S_];        // temp F*P
  __shared__ float HP[M_ * S_];        // H*Ppred, later X = Sm^{-1} H Ppred
  __shared__ float HP0[M_ * S_];       // pristine copy of H*Ppred
  __shared__ float Sm[M_ * (M_ + 1)];  // innovation covariance (padded)

  const int tid = threadIdx.x;

  for (int i = tid; i < S_ * S_; i += 256) P[i] = P0[i];
  __syncthreads();

  for (int t = 0; t < T_; ++t) {
    // Pp = F * P
    for (int e = tid; e < S_ * S_; e += 256) {
      int i = e >> 6, j = e & 63;
      float acc = 0.f;
      for (int k = 0; k < S_; ++k) acc += F[i * S_ + k] * P[k * S_ + j];
      Pp[e] = acc;
    }
    __syncthreads();
    // P = Pp * F^T + Q   (P becomes P_pred)
    for (int e = tid; e < S_ * S_; e += 256) {
      int i = e >> 6, j = e & 63;
      float acc = Q[e];
      for (int k = 0; k < S_; ++k) acc += Pp[i * S_ + k] * F[j * S_ + k];
      P[e] = acc;
    }
    __syncthreads();
    // HP = H * P_pred (keep pristine copy HP0)
    for (int e = tid; e < M_ * S_; e += 256) {
      int m = e >> 6, j = e & 63;
      float acc = 0.f;
      for (int k = 0; k < S_; ++k) acc += H[m * S_ + k] * P[k * S_ + j];
      HP[e] = acc;
      HP0[e] = acc;
    }
    __syncthreads();
    // Sm = HP * H^T + R
    for (int e = tid; e < M_ * M_; e += 256) {
      int m = e >> 5, n = e & 31;
      float acc = R[e];
      for (int k = 0; k < S_; ++k) acc += HP[m * S_ + k] * H[n * S_ + k];
      Sm[m * (M_ + 1) + n] = acc;
    }
    __syncthreads();

    // Gauss-Jordan (Sm is SPD, no pivoting): HP <- Sm^{-1} * HP
    for (int col = 0; col < M_; ++col) {
      const float rp = 1.0f / Sm[col * (M_ + 1) + col];
      __syncthreads();
      for (int j = tid; j < 96; j += 256) {  // scale pivot row
        if (j < M_)
          Sm[col * (M_ + 1) + j] *= rp;
        else
          HP[col * S_ + (j - M_)] *= rp;
      }
      __syncthreads();
      for (int idx = tid; idx < M_ * 96; idx += 256) {  // eliminate
        int r = idx / 96, j = idx % 96;
        if (r != col) {
          float f = Sm[r * (M_ + 1) + col];
          if (j < M_) {
            if (j != col) Sm[r * (M_ + 1) + j] -= f * Sm[col * (M_ + 1) + j];
          } else {
            HP[r * S_ + (j - M_)] -= f * HP[col * S_ + (j - M_)];
          }
        }
      }
      __syncthreads();
      if (tid < M_ && tid != col) Sm[tid * (M_ + 1) + col] = 0.f;
      __syncthreads();
    }

    // K_t = X^T  (X = HP after solve) ; store row-major [S_][M_]
    float* Kt = Kall + (size_t)t * (S_ * M_);
    for (int e = tid; e < S_ * M_; e += 256) {
      int s = e >> 5, m = e & 31;
      Kt[e] = HP[m * S_ + s];
    }
    // P = P_pred - K * HP0 :  P[i][j] -= sum_m X[m][i] * HP0[m][j]
    for (int e = tid; e < S_ * S_; e += 256) {
      int i = e >> 6, j = e & 63;
      float acc = 0.f;
      for (int m = 0; m < M_; ++m) acc += HP[m * S_ + i] * HP0[m * S_ + j];
      P[e] -= acc;
    }
    __syncthreads();
  }
}

// ---------------------------------------------------------------------------
// Kernel 2: mean recursion. 16 batches/block = N dim of fp32 WMMA tiles,
// 4 waves split the M row-tiles. F/H A-fragments live in VGPRs; K_t is
// double-buffered in LDS by the Tensor Data Mover one step ahead.
// ---------------------------------------------------------------------------
__global__ __launch_bounds__(128) void kf_means(
    const float* __restrict__ obs, const float* __restrict__ F,
    const float* __restrict__ H, const float* __restrict__ x0,
    const float* __restrict__ Kall, float* __restrict__ out) {
  __shared__ float Xs[S_ * 16];         // predicted state [row][batch-col]
  __shared__ float Xu[S_ * 16];         // updated state
  __shared__ float Rs[M_ * 16];         // residual
  __shared__ float Kbuf[2][S_ * M_];    // TDM double buffer for K_t

  const int tid = threadIdx.x;
  const int wave = tid >> 5;
  const int lane = tid & 31;
  const int col = lane & 15;  // N index (local batch)
  const int hi = lane >> 4;   // lane-half selector
  const int bat = blockIdx.x * 16 + col;
  const int mt = wave;        // row tile owned by this wave

  // Loop-invariant A-fragments: F (all waves), H (waves 0,1)
  v2f ff[16];
#pragma unroll
  for (int kt = 0; kt < 16; ++kt)
    ff[kt] = *(const v2f*)&F[(mt * 16 + col) * S_ + kt * 4 + 2 * hi];
  v2f hf[16];
  if (wave < 2) {
#pragma unroll
    for (int kt = 0; kt < 16; ++kt)
      hf[kt] = *(const v2f*)&H[(wave * 16 + col) * S_ + kt * 4 + 2 * hi];
  }

  // init: x_pred(0) = F * x0, broadcast across the 16 batch columns
  if (tid < S_) {
    float acc = 0.f;
    for (int k = 0; k < S_; ++k) acc += F[tid * S_ + k] * x0[k];
    for (int c = 0; c < 16; ++c) Xs[tid * 16 + c] = acc;
  }
  // stage K_0 into buffer 0 (one DMA from wave 0; EXEC-agnostic instruction)
  if (wave == 0)
    tdm_copy_gain(Kall, (unsigned)(uintptr_t)(void*)&Kbuf[0][0]);
  __syncthreads();

  for (int t = 0; t < T_; ++t) {
    const int cur = t & 1;
    // kick off DMA of next step's gain while this step computes
    if (wave == 0 && t + 1 < T_)
      tdm_copy_gain(Kall + (size_t)(t + 1) * (S_ * M_),
                    (unsigned)(uintptr_t)(void*)&Kbuf[cur ^ 1][0]);

    // ---- Phase A (waves 0,1): Y = H * Xs ; out = Y ; Rs = obs - Y ----
    if (wave < 2) {
      const float* obs_t = obs + ((size_t)bat * T_ + t) * M_;
      float* out_t = out + ((size_t)bat * T_ + t) * M_;
      float ov[8];
#pragma unroll
      for (int r = 0; r < 8; ++r) ov[r] = obs_t[wave * 16 + hi * 8 + r];
      v2f bf[16];
#pragma unroll
      for (int kt = 0; kt < 16; ++kt) {
        bf[kt].x = Xs[(kt * 4 + 2 * hi + 0) * 16 + col];
        bf[kt].y = Xs[(kt * 4 + 2 * hi + 1) * 16 + col];
      }
      v8f acc = {};
#pragma unroll
      for (int kt = 0; kt < 16; ++kt) acc = wmma_f32_16x16x4(hf[kt], bf[kt], acc);
#pragma unroll
      for (int r = 0; r < 8; ++r) {
        int m = wave * 16 + hi * 8 + r;
        out_t[m] = acc[r];                 // one-step-ahead prediction
        Rs[m * 16 + col] = ov[r] - acc[r]; // residual
      }
    } else if (t + 1 < T_) {
      // idle waves pull next obs rows toward the caches
      __builtin_prefetch(
          obs + ((size_t)bat * T_ + (t + 1)) * M_ + (wave - 2) * 16, 0, 0);
    }
    // ensure Kbuf[cur] landed (<=1 outstanding keeps next DMA in flight)
    if (wave == 0) {
      if (t + 1 < T_)
        __builtin_amdgcn_s_wait_tensorcnt((short)1);
      else
        __builtin_amdgcn_s_wait_tensorcnt((short)0);
    }
    __syncthreads();

    // ---- Phase B (all waves): Xu = Xs + K_t * Rs ----
    {
      const float* Kb = &Kbuf[cur][0];
      v8f acc;
#pragma unroll
      for (int r = 0; r < 8; ++r)
        acc[r] = Xs[(mt * 16 + hi * 8 + r) * 16 + col];
      v2f ka[8], bf[8];
#pragma unroll
      for (int kt = 0; kt < 8; ++kt) {
        ka[kt] = *(const v2f*)&Kb[(mt * 16 + col) * M_ + kt * 4 + 2 * hi];
        bf[kt].x = Rs[(kt * 4 + 2 * hi + 0) * 16 + col];
        bf[kt].y = Rs[(kt * 4 + 2 * hi + 1) * 16 + col];
      }
#pragma unroll
      for (int kt = 0; kt < 8; ++kt) acc = wmma_f32_16x16x4(ka[kt], bf[kt], acc);
#pragma unroll
      for (int r = 0; r < 8; ++r)
        Xu[(mt * 16 + hi * 8 + r) * 16 + col] = acc[r];
    }
    __syncthreads();

    // ---- Phase C (all waves): Xs = F * Xu ----
    {
      v2f bf[16];
#pragma unroll
      for (int kt = 0; kt < 16; ++kt) {
        bf[kt].x = Xu[(kt * 4 + 2 * hi + 0) * 16 + col];
        bf[kt].y = Xu[(kt * 4 + 2 * hi + 1) * 16 + col];
      }
      v8f acc = {};
#pragma unroll
      for (int kt = 0; kt < 16; ++kt) acc = wmma_f32_16x16x4(ff[kt], bf[kt], acc);
#pragma unroll
      for (int r = 0; r < 8; ++r)
        Xs[(mt * 16 + hi * 8 + r) * 16 + col] = acc[r];
    }
    __syncthreads();
  }
}

// ---------------------------------------------------------------------------
extern "C" void kernel_launch(void* const* d_in, const int* in_sizes, int n_in,
                              void* d_out, int out_size, void* d_ws,
                              size_t ws_size, hipStream_t stream) {
  const float* obs = (const float*)d_in[0];  // [B,T,M]
  const float* F = (const float*)d_in[1];    // [S,S]
  const float* H = (const float*)d_in[2];    // [M,S]
  const float* Q = (const float*)d_in[3];    // [S,S]
  const float* R = (const float*)d_in[4];    // [M,M]
  const float* x0 = (const float*)d_in[5];   // [S]
  const float* P0 = (const float*)d_in[6];   // [S,S]
  float* out = (float*)d_out;                // [B,T,M]
  float* Kall = (float*)d_ws;                // [T,S,M] = 4 MB scratch

  kf_gains<<<1, 256, 0, stream>>>(F, H, Q, R, P0, Kall);
  kf_means<<<B_ / 16, 128, 0, stream>>>(obs, F, H, x0, Kall, out);
}